// DCPLoss_31653908972140
// MI455X (gfx1250) — compile-verified
//
#include <hip/hip_runtime.h>
#include <hip/hip_bf16.h>

#define Bb 8
#define Hh 256
#define Ww 256
#define HW (Hh*Ww)
#define PADW 7
#define KTOP 65
#define PH (Hh-2)
#define PW (Ww-2)
#define NPATCH (PH*PW)
#define TOTPATCH (Bb*NPATCH)
#define EPS_ 1e-5f

typedef __attribute__((ext_vector_type(2))) float v2f;
typedef __attribute__((ext_vector_type(8))) float v8f;

// full-wave (32-lane) xor reduction: result valid in all lanes
__device__ __forceinline__ float wred(float v) {
    #pragma unroll
    for (int o = 16; o > 0; o >>= 1) v += __shfl_xor(v, o, 32);
    return v;
}

// half-wave reduction: sums lanes 0-15 (lanes 16-31 reduce separately to their
// own group). Sufficient when inputs live in lanes 0-8 and consumers are lanes 0-15.
__device__ __forceinline__ float wred16(float v) {
    #pragma unroll
    for (int o = 8; o > 0; o >>= 1) v += __shfl_xor(v, o, 32);
    return v;
}

__global__ void k_init(float* acc) {
    if (threadIdx.x < 2) acc[threadIdx.x] = 0.0f;
}

// per-pixel min over the 3 channels
__global__ void k_chmin(const float* __restrict__ img, float* __restrict__ out) {
    int tot = Bb * HW;
    for (int i = blockIdx.x * blockDim.x + threadIdx.x; i < tot; i += gridDim.x * blockDim.x) {
        int b = i / HW, p = i % HW;
        const float* base = img + (size_t)b * 3 * HW + p;
        out[i] = fminf(base[0], fminf(base[HW], base[2 * HW]));
    }
}

// horizontal 15-wide min (boundary = +inf pad, i.e. skip)
__global__ void k_rowmin(const float* __restrict__ in, float* __restrict__ out) {
    int tot = Bb * HW;
    for (int i = blockIdx.x * blockDim.x + threadIdx.x; i < tot; i += gridDim.x * blockDim.x) {
        int rem = i % HW;
        int w = rem % Ww;
        int lo = (w - PADW < 0) ? 0 : w - PADW;
        int hi = (w + PADW > Ww - 1) ? Ww - 1 : w + PADW;
        const float* row = in + (i - w);
        float m = row[lo];
        for (int x = lo + 1; x <= hi; ++x) m = fminf(m, row[x]);
        out[i] = m;
    }
}

// vertical 15-wide min
__global__ void k_colmin(const float* __restrict__ in, float* __restrict__ out) {
    int tot = Bb * HW;
    for (int i = blockIdx.x * blockDim.x + threadIdx.x; i < tot; i += gridDim.x * blockDim.x) {
        int rem = i % HW;
        int h = rem / Ww, w = rem % Ww;
        int base = i - rem;
        int lo = (h - PADW < 0) ? 0 : h - PADW;
        int hi = (h + PADW > Hh - 1) ? Hh - 1 : h + PADW;
        float m = in[base + lo * Ww + w];
        for (int y = lo + 1; y <= hi; ++y) m = fminf(m, in[base + y * Ww + w]);
        out[i] = m;
    }
}

// per-batch: bisection for 65th-largest dark-channel value, then argmax of
// max-channel intensity over candidates; A[b][c] = img at that pixel.
__global__ void k_atmos(const float* __restrict__ img, const float* __restrict__ dc,
                        float* __restrict__ A) {
    int b = blockIdx.x;
    const float* d = dc + (size_t)b * HW;
    __shared__ unsigned sc;
    __shared__ float sval[256];
    __shared__ int   sidx[256];
    unsigned prefix = 0;
    for (int bit = 30; bit >= 0; --bit) {
        unsigned cand = prefix | (1u << bit);
        if (threadIdx.x == 0) sc = 0;
        __syncthreads();
        unsigned local = 0;
        for (int p = threadIdx.x; p < HW; p += blockDim.x)
            if (__float_as_uint(d[p]) >= cand) local++;
        atomicAdd(&sc, local);
        __syncthreads();
        if (sc >= KTOP) prefix = cand;
        __syncthreads();
    }
    float bestv = -1.0f; int besti = 0;
    for (int p = threadIdx.x; p < HW; p += blockDim.x) {
        if (__float_as_uint(d[p]) >= prefix) {
            const float* base = img + (size_t)b * 3 * HW + p;
            float mx = fmaxf(base[0], fmaxf(base[HW], base[2 * HW]));
            if (mx > bestv) { bestv = mx; besti = p; }
        }
    }
    sval[threadIdx.x] = bestv; sidx[threadIdx.x] = besti;
    __syncthreads();
    for (int s = blockDim.x / 2; s > 0; s >>= 1) {
        if (threadIdx.x < s) {
            float ov = sval[threadIdx.x + s]; int oi = sidx[threadIdx.x + s];
            if (ov > sval[threadIdx.x] || (ov == sval[threadIdx.x] && oi < sidx[threadIdx.x])) {
                sval[threadIdx.x] = ov; sidx[threadIdx.x] = oi;
            }
        }
        __syncthreads();
    }
    if (threadIdx.x < 3)
        A[b * 4 + threadIdx.x] = img[(size_t)b * 3 * HW + (size_t)threadIdx.x * HW + sidx[0]];
}

// min over channels of img / A
__global__ void k_normmin(const float* __restrict__ img, const float* __restrict__ A,
                          float* __restrict__ out) {
    int tot = Bb * HW;
    for (int i = blockIdx.x * blockDim.x + threadIdx.x; i < tot; i += gridDim.x * blockDim.x) {
        int b = i / HW, p = i % HW;
        const float* base = img + (size_t)b * 3 * HW + p;
        float r0 = base[0]      / A[b * 4 + 0];
        float r1 = base[HW]     / A[b * 4 + 1];
        float r2 = base[2 * HW] / A[b * 4 + 2];
        out[i] = fminf(r0, fminf(r1, r2));
    }
}

// prior = sum (y - (1 - 0.95*dcn))^2
__global__ void k_prior(const float* __restrict__ y, const float* __restrict__ dcn,
                        float* __restrict__ acc) {
    int tot = Bb * HW;
    float s = 0.0f;
    for (int i = blockIdx.x * blockDim.x + threadIdx.x; i < tot; i += gridDim.x * blockDim.x) {
        float t = 1.0f - 0.95f * dcn[i];
        float df = y[i] - t;
        s += df * df;
    }
    s = wred(s);
    if ((threadIdx.x & 31) == 0) atomicAdd(acc + 1, s);
}

// Matting-Laplacian fidelity: one wave per 3x3 patch, the 9x9 weights Gram
// (q @ Xc^T) is computed in a single V_WMMA_F32_16X16X4_F32.
__global__ void k_fid(const float* __restrict__ img, const float* __restrict__ y,
                      float* __restrict__ acc) {
    const int lane = threadIdx.x & 31;
    const int wid  = (blockIdx.x * blockDim.x + threadIdx.x) >> 5;
    const int nw   = (gridDim.x * blockDim.x) >> 5;
    const float inv9 = 1.0f / 9.0f;
    float faccum = 0.0f;

    for (int pidx = wid; pidx < TOTPATCH; pidx += nw) {   // wave-uniform loop
        int b  = pidx / NPATCH;
        int n  = pidx % NPATCH;
        int py = n / PW, px = n % PW;

        float x0 = 0.f, x1 = 0.f, x2 = 0.f, yv = 0.f;
        if (lane < 9) {
            int dy = lane / 3, dx = lane % 3;
            size_t off = (size_t)b * 3 * HW + (size_t)(py + dy) * Ww + (px + dx);
            x0 = img[off]; x1 = img[off + HW]; x2 = img[off + 2 * HW];
            yv = y[(size_t)b * HW + (size_t)(py + dy) * Ww + (px + dx)];
        }

        // ndsum = sum_{ij}(y_i - y_j)^2 = 18*Sum y^2 - 2*(Sum y)^2
        // inputs live in lanes 0-8; results only consumed by lanes 0-15 -> wred16
        float sy  = wred16(yv);
        float sy2 = wred16(yv * yv);
        float ndsum = 18.0f * sy2 - 2.0f * sy * sy;

        // 3x3 covariance of patch colors (+ eps/9 on the diagonal)
        float s0  = wred16(x0),      s1  = wred16(x1),      s2  = wred16(x2);
        float s00 = wred16(x0 * x0), s01 = wred16(x0 * x1), s02 = wred16(x0 * x2);
        float s11 = wred16(x1 * x1), s12 = wred16(x1 * x2), s22 = wred16(x2 * x2);
        float m0 = s0 * inv9, m1 = s1 * inv9, m2 = s2 * inv9;
        float a  = s00 * inv9 - m0 * m0 + EPS_ * inv9;
        float bq = s01 * inv9 - m0 * m1;
        float cq = s02 * inv9 - m0 * m2;
        float dq = s11 * inv9 - m1 * m1 + EPS_ * inv9;
        float e  = s12 * inv9 - m1 * m2;
        float f  = s22 * inv9 - m2 * m2 + EPS_ * inv9;
        // symmetric 3x3 inverse via adjugate
        float A00 = dq * f - e * e,  A01 = cq * e - bq * f, A02 = bq * e - cq * dq;
        float A11 = a * f - cq * cq, A12 = bq * cq - a * e, A22 = a * dq - bq * bq;
        float rdet = 1.0f / (a * A00 + bq * A01 + cq * A02);
        A00 *= rdet; A01 *= rdet; A02 *= rdet; A11 *= rdet; A12 *= rdet; A22 *= rdet;

        float xc0 = x0 - m0, xc1 = x1 - m1, xc2 = x2 - m2;
        if (lane >= 9) { xc0 = 0.f; xc1 = 0.f; xc2 = 0.f; }
        float q0 = xc0 * A00 + xc1 * A01 + xc2 * A02;
        float q1 = xc0 * A01 + xc1 * A11 + xc2 * A12;
        float q2 = xc0 * A02 + xc1 * A12 + xc2 * A22;

        // Assemble WMMA 16x16x4 f32 operands:
        // A (16x4): lanes 0-15 -> {A[m][0],A[m][1]}, lanes 16-31 -> {A[m][2],A[m][3]=0}
        // B (4x16): lanes 0-15 -> {B[0][n],B[1][n]}, lanes 16-31 -> {B[2][n],B[3][n]=0}
        int  m  = lane & 15;
        bool lo = lane < 16;
        float q2s  = __shfl(q2,  m, 32);
        float xc2s = __shfl(xc2, m, 32);
        v2f av, bv;
        av.x = lo ? q0  : q2s;   av.y = lo ? q1  : 0.0f;
        bv.x = lo ? xc0 : xc2s;  bv.y = lo ? xc1 : 0.0f;
        v8f cz = {};
        v8f dmat = __builtin_amdgcn_wmma_f32_16x16x4_f32(
            false, av, false, bv, (short)0, cz, false, false);

        // sum the valid 9x9 block of D (entries live in lanes 0-8 and 16-24,
        // so this reduction must span the full wave)
        float part;
        if (lane < 9) {
            part = dmat[0] + dmat[1] + dmat[2] + dmat[3] +
                   dmat[4] + dmat[5] + dmat[6] + dmat[7];   // M=0..7, N=lane
        } else if (lane >= 16 && lane < 25) {
            part = dmat[0];                                  // M=8, N=lane-16
        } else {
            part = 0.0f;
        }
        float S = wred(part);
        float wsum = (S + 81.0f) * inv9;
        faccum += ndsum * wsum;
    }
    if (lane == 0) atomicAdd(acc, faccum);
}

__global__ void k_final(const float* __restrict__ acc, float* __restrict__ out) {
    if (blockIdx.x == 0 && threadIdx.x == 0) {
        // loss = (LAM1*0.5*fid + LAM2*prior) / N, LAM1=2, LAM2=0.01
        out[0] = (acc[0] + 0.01f * acc[1]) * (1.0f / (float)NPATCH);
    }
}

extern "C" void kernel_launch(void* const* d_in, const int* in_sizes, int n_in,
                              void* d_out, int out_size, void* d_ws, size_t ws_size,
                              hipStream_t stream) {
    const float* img = (const float*)d_in[0];   // [8,3,256,256]
    const float* yp  = (const float*)d_in[1];   // [8,1,256,256]
    float* out = (float*)d_out;

    float* bufA = (float*)d_ws;                 // B*H*W
    float* bufB = bufA + (size_t)Bb * HW;       // B*H*W
    float* Aat  = bufB + (size_t)Bb * HW;       // 8*4
    float* acc  = Aat + Bb * 4;                 // [fid, prior]

    dim3 blk(256);
    dim3 grdE(1024);
    dim3 grdP(2048);

    k_init<<<1, 32, 0, stream>>>(acc);
    k_chmin  <<<grdE, blk, 0, stream>>>(img, bufA);
    k_rowmin <<<grdE, blk, 0, stream>>>(bufA, bufB);
    k_colmin <<<grdE, blk, 0, stream>>>(bufB, bufA);          // dc in bufA
    k_atmos  <<<Bb, 256, 0, stream>>>(img, bufA, Aat);
    k_normmin<<<grdE, blk, 0, stream>>>(img, Aat, bufB);
    k_rowmin <<<grdE, blk, 0, stream>>>(bufB, bufA);
    k_colmin <<<grdE, blk, 0, stream>>>(bufA, bufB);          // dcn in bufB
    k_prior  <<<grdE, blk, 0, stream>>>(yp, bufB, acc);
    k_fid    <<<grdP, blk, 0, stream>>>(img, yp, acc);
    k_final  <<<1, 32, 0, stream>>>(acc, out);
}